// BoundaryAwareBCELoss_10642928959650
// MI455X (gfx1250) — compile-verified
//
#include <hip/hip_runtime.h>
#include <stdint.h>

// Boundary-aware BCE loss for B=32, H=W=1024 fp32 inputs.
// Memory-bound: 268 MB of reads -> ~11.5us floor at 23.3 TB/s.
// Strategy: async global->LDS tile load of `target` (9x reuse in the 3x3
// stencil), streamed coalesced loads of `pred`, separable rolling min/max,
// one fast log per pixel, deterministic two-pass reduction.

#define TILE_W   64
#define TILE_H   32
#define LDSW     72          // 4 halo + 64 + 4 halo (16B-aligned chunks)
#define LDSROWS  34          // 1 halo + 32 + 1 halo
#define NTHREADS 256

typedef int v4i __attribute__((vector_size(4 * sizeof(int))));
typedef __attribute__((address_space(1))) v4i* g_v4i_p;   // global (device)
typedef __attribute__((address_space(3))) v4i* l_v4i_p;   // LDS
typedef __attribute__((address_space(3))) void* as3_vp;

__device__ __forceinline__ void async_copy_b128(const float* gp, void* lp) {
#if __has_builtin(__builtin_amdgcn_global_load_async_to_lds_b128)
  __builtin_amdgcn_global_load_async_to_lds_b128(
      (g_v4i_p)(uintptr_t)gp, (l_v4i_p)lp, 0, 0);
#else
  uint32_t loff = (uint32_t)(uintptr_t)(as3_vp)lp;
  asm volatile("global_load_async_to_lds_b128 %0, %1, off"
               :
               : "v"(loff), "v"(gp)
               : "memory");
#endif
}

__device__ __forceinline__ void wait_async0() {
#if __has_builtin(__builtin_amdgcn_s_wait_asynccnt)
  __builtin_amdgcn_s_wait_asynccnt(0);
#else
  asm volatile("s_wait_asynccnt 0" ::: "memory");
#endif
  asm volatile("" ::: "memory");
}

__global__ __launch_bounds__(NTHREADS)
void babce_main(const float* __restrict__ pred,
                const float* __restrict__ target,
                float* __restrict__ partials,
                int Himg, int Wimg)
{
  __shared__ float tile[LDSROWS][LDSW];
  __shared__ float red[NTHREADS];

  const int tid = threadIdx.x;
  const int gx0 = blockIdx.x * TILE_W;
  const int gy0 = blockIdx.y * TILE_H;
  const size_t plane = (size_t)Himg * (size_t)Wimg;
  const float* tbase = target + (size_t)blockIdx.z * plane;
  const float* pbase = pred   + (size_t)blockIdx.z * plane;

  // ---- Phase 1: async-load target tile (+halo) into LDS, 16B chunks ----
  const int CPR = LDSW / 4;            // 18 chunks per row
  const int CHUNKS = LDSROWS * CPR;    // 612
  for (int i = tid; i < CHUNKS; i += NTHREADS) {
    int r = i / CPR;
    int c = i - r * CPR;
    int grow = gy0 - 1 + r;                 // clamp-to-edge == ref's +/-inf pad
    grow = grow < 0 ? 0 : (grow > Himg - 1 ? Himg - 1 : grow);
    int gcol = gx0 - 4 + c * 4;             // 16B-aligned chunk start
    gcol = gcol < 0 ? 0 : (gcol > Wimg - 4 ? Wimg - 4 : gcol);
    async_copy_b128(tbase + (size_t)grow * Wimg + gcol, &tile[r][c * 4]);
  }
  wait_async0();
  __syncthreads();

  // x-halo replicate patch for image-edge blocks (block-uniform branches)
  if (gx0 == 0) {
    if (tid < LDSROWS) tile[tid][3] = tile[tid][4];
  }
  if (gx0 + TILE_W == Wimg) {
    if (tid < LDSROWS) tile[tid][4 + TILE_W] = tile[tid][3 + TILE_W];
  }
  if (gx0 == 0 || gx0 + TILE_W == Wimg) __syncthreads();

  // ---- Phase 2: each thread = 1 column x 8 rows, rolling 3x3 min/max ----
  const int col = tid & (TILE_W - 1);
  const int rb  = (tid >> 6) * 8;      // local row base: 0,8,16,24
  const int lc  = 4 + col;

  const float* pp = pbase + (size_t)(gy0 + rb) * Wimg + (gx0 + col);
  float pv[8];
#pragma unroll
  for (int k = 0; k < 8; ++k) pv[k] = pp[(size_t)k * Wimg];

  float mn0, mx0, mn1, mx1, md1;
  {
    const float* r0 = &tile[rb][lc];
    float a = r0[-1], b = r0[0], c = r0[1];
    mn0 = fminf(a, fminf(b, c)); mx0 = fmaxf(a, fmaxf(b, c));
    const float* r1 = &tile[rb + 1][lc];
    a = r1[-1]; b = r1[0]; c = r1[1];
    mn1 = fminf(a, fminf(b, c)); mx1 = fmaxf(a, fmaxf(b, c)); md1 = b;
  }

  float acc = 0.0f;
#pragma unroll
  for (int k = 0; k < 8; ++k) {
    const float* r2 = &tile[rb + 2 + k][lc];
    float a = r2[-1], b = r2[0], c = r2[1];
    float mn2 = fminf(a, fminf(b, c));
    float mx2 = fmaxf(a, fmaxf(b, c));
    float wmax = fmaxf(mx0, fmaxf(mx1, mx2));
    float wmin = fminf(mn0, fminf(mn1, mn2));
    float t = md1;                 // center target value (binary 0/1)
    float p = pv[k];
    float x = t > 0.5f ? p : (1.0f - p);      // bce = -log(x)
    float w = (wmax - wmin) > 0.5f ? 3.0f : 1.0f;
    acc = fmaf(w, -__logf(x), acc);
    mn0 = mn1; mx0 = mx1;
    mn1 = mn2; mx1 = mx2; md1 = b;
  }

  // ---- Phase 3: block reduction -> per-block partial ----
  red[tid] = acc;
  __syncthreads();
#pragma unroll
  for (int s = NTHREADS / 2; s > 0; s >>= 1) {
    if (tid < s) red[tid] += red[tid + s];
    __syncthreads();
  }
  if (tid == 0) {
    int bidx = (blockIdx.z * gridDim.y + blockIdx.y) * gridDim.x + blockIdx.x;
    partials[bidx] = red[0];
  }
}

__global__ __launch_bounds__(NTHREADS)
void babce_reduce(const float* __restrict__ partials, int n,
                  float* __restrict__ out, double inv_n)
{
  __shared__ double red[NTHREADS];
  const int tid = threadIdx.x;
  double a = 0.0;
  for (int i = tid; i < n; i += NTHREADS) a += (double)partials[i];
  red[tid] = a;
  __syncthreads();
#pragma unroll
  for (int s = NTHREADS / 2; s > 0; s >>= 1) {
    if (tid < s) red[tid] += red[tid + s];
    __syncthreads();
  }
  if (tid == 0) out[0] = (float)(red[0] * inv_n);
}

extern "C" void kernel_launch(void* const* d_in, const int* in_sizes, int n_in,
                              void* d_out, int out_size, void* d_ws, size_t ws_size,
                              hipStream_t stream) {
  (void)n_in; (void)out_size; (void)ws_size;
  const float* pred   = (const float*)d_in[0];
  const float* target = (const float*)d_in[1];
  float* out          = (float*)d_out;
  float* partials     = (float*)d_ws;

  const int H = 1024, W = 1024;
  const int B = in_sizes[0] / (H * W);          // 32 for the reference shapes

  dim3 grid(W / TILE_W, H / TILE_H, B);         // (16, 32, 32) = 16384 blocks
  babce_main<<<grid, NTHREADS, 0, stream>>>(pred, target, partials, H, W);

  const int nblocks = (W / TILE_W) * (H / TILE_H) * B;
  const double inv_n = 1.0 / ((double)B * (double)H * (double)W);
  babce_reduce<<<1, NTHREADS, 0, stream>>>(partials, nblocks, out, inv_n);
}